// DynamicTopK_71296457114397
// MI455X (gfx1250) — compile-verified
//
#include <hip/hip_runtime.h>

// DynamicTopK (softmax -> top-p/top-k mask -> gather) for MI455X / gfx1250.
//
// Per row (1024 f32): out[:] = -inf except out[rank(s)] = prob_sorted[s] for
// original columns s < m, where m = min(first cumsum>0.6 position + 1, 8).
//
// HBM-bound: 537 MB total traffic / 23.3 TB/s ~ 23 us floor. One 256-thread
// block per row (grid-stride). CDNA5 paths used:
//   * global_load_async_to_lds_b128 + s_wait_asynccnt: double-buffered row
//     staging through LDS (ASYNCcnt pipeline, 8 KB LDS / block).
//   * v_wmma_f32_16x16x4_f32 with B=ones: 128-wide exp-sum reduction per wave
//     in 2 matrix ops (D = A x 1 replicates row sums; +1 shfl finishes).
//   * ballot+popcount ranks ride the SALU alongside VALU work.

#define N_COLS   1024
#define TOPK     8
#define TOPP_F   0.6f
#define NBLK_THR 256
#define NWAVES   (NBLK_THR / 32)

#if defined(__HIP_DEVICE_COMPILE__) && __has_builtin(__builtin_amdgcn_wmma_f32_16x16x4_f32)
#define USE_WMMA_SUM 1
#endif

typedef __attribute__((ext_vector_type(2))) float v2f;
typedef __attribute__((ext_vector_type(8))) float v8f;

__device__ __forceinline__ void ceD(float& a, float& b) {  // desc compare-exchange
  float mx = fmaxf(a, b), mn = fminf(a, b);
  a = mx; b = mn;
}

// Sort a bitonic 8-sequence into descending order (3 stages x 4 CE).
__device__ __forceinline__ void bitonic8(float t[8]) {
  ceD(t[0],t[4]); ceD(t[1],t[5]); ceD(t[2],t[6]); ceD(t[3],t[7]);
  ceD(t[0],t[2]); ceD(t[1],t[3]); ceD(t[4],t[6]); ceD(t[5],t[7]);
  ceD(t[0],t[1]); ceD(t[2],t[3]); ceD(t[4],t[5]); ceD(t[6],t[7]);
}

__global__ __launch_bounds__(NBLK_THR)
void topk_topp_kernel(const float* __restrict__ x, float* __restrict__ out, int rows)
{
  __shared__ __align__(16) float rowbuf[2][N_COLS];   // double-buffered staged row
  __shared__ __align__(16) float maxBuf[NWAVES];
  __shared__ __align__(16) float sumBuf[NWAVES];
  __shared__ __align__(16) int   cntBuf[NWAVES * 8];  // per-wave rank partial counts
  __shared__ __align__(16) float topBuf[NWAVES * 8];  // per-wave sorted top-8 (exp values)

  const int tid  = threadIdx.x;
  const int lane = tid & 31;
  const int wv   = tid >> 5;
  const float NEG_INF = __int_as_float(0xff800000);

  int cur = 0;
  const int row0   = blockIdx.x;
  const int stride = gridDim.x;

  // Per-wave async DMA of its 512B slice: GVS mode (SGPR base + per-lane voffset).
  #define ISSUE_ASYNC(BUF, R) do {                                              \
      const void* _src = (const void*)(x + (size_t)(R) * N_COLS);               \
      unsigned _lds = (unsigned)(unsigned long long)(&rowbuf[(BUF)][0])         \
                      + (unsigned)(tid * 16);                                   \
      unsigned _go  = (unsigned)(tid * 16);                                     \
      asm volatile("global_load_async_to_lds_b128 %0, %1, %2"                   \
                   :: "v"(_lds), "v"(_go), "s"(_src) : "memory");               \
    } while (0)

  if (row0 < rows) ISSUE_ASYNC(0, row0);

  for (int row = row0; row < rows; row += stride) {
    const int nrow = row + stride;
    if (nrow < rows) {
      ISSUE_ASYNC(cur ^ 1, nrow);                       // prefetch next row
      asm volatile("s_wait_asynccnt 0x1" ::: "memory"); // current row done, 1 in flight
    } else {
      asm volatile("s_wait_asynccnt 0x0" ::: "memory");
    }
    __syncthreads();                                    // (A) staged row visible

    const float4 xv = ((const float4*)&rowbuf[cur][0])[tid]; // my 4 columns
    const float4 qa = ((const float4*)&rowbuf[cur][0])[0];   // cols 0..3 (broadcast)
    const float4 qb = ((const float4*)&rowbuf[cur][0])[1];   // cols 4..7 (broadcast)

    // ---------- block max ----------
    float mx = fmaxf(fmaxf(xv.x, xv.y), fmaxf(xv.z, xv.w));
    #pragma unroll
    for (int d = 16; d >= 1; d >>= 1) mx = fmaxf(mx, __shfl_xor(mx, d, 32));
    if (lane == 0) maxBuf[wv] = mx;
    __syncthreads();                                    // (B)
    {
      const float4 m0 = ((const float4*)maxBuf)[0];
      const float4 m1 = ((const float4*)maxBuf)[1];
      mx = fmaxf(fmaxf(fmaxf(m0.x, m0.y), fmaxf(m0.z, m0.w)),
                 fmaxf(fmaxf(m1.x, m1.y), fmaxf(m1.z, m1.w)));
    }

    // ---------- exp ----------
    float4 e;
    e.x = __expf(xv.x - mx); e.y = __expf(xv.y - mx);
    e.z = __expf(xv.z - mx); e.w = __expf(xv.w - mx);

    // ---------- wave exp-sum: 2x v_wmma_f32_16x16x4_f32 with B = ones ----------
    float ws;
#ifdef USE_WMMA_SUM
    {
      v2f a01  = {e.x, e.y};
      v2f a23  = {e.z, e.w};
      v2f ones = {1.0f, 1.0f};
      v8f acc  = {0, 0, 0, 0, 0, 0, 0, 0};
      acc = __builtin_amdgcn_wmma_f32_16x16x4_f32(false, a01, false, ones, (short)0, acc, false, false);
      acc = __builtin_amdgcn_wmma_f32_16x16x4_f32(false, a23, false, ones, (short)0, acc, false, false);
      // lane holds 8 row-sums (half of the 16 rows); one xor-16 completes the 128-sum
      float h = ((acc[0] + acc[1]) + (acc[2] + acc[3])) +
                ((acc[4] + acc[5]) + (acc[6] + acc[7]));
      ws = h + __shfl_xor(h, 16, 32);
    }
#else
    {
      float h = (e.x + e.y) + (e.z + e.w);
      #pragma unroll
      for (int d = 16; d >= 1; d >>= 1) h += __shfl_xor(h, d, 32);
      ws = h;
    }
#endif
    if (lane == 0) sumBuf[wv] = ws;

    // ---------- ranks of original columns 0..7 (ballot + SALU popcount) ----------
    {
      const float xs[8] = {qa.x, qa.y, qa.z, qa.w, qb.x, qb.y, qb.z, qb.w};
      int c[8];
      #pragma unroll
      for (int s = 0; s < 8; ++s) {
        int cc = 0;
        cc += __popcll(__ballot(xv.x > xs[s]));
        cc += __popcll(__ballot(xv.y > xs[s]));
        cc += __popcll(__ballot(xv.z > xs[s]));
        cc += __popcll(__ballot(xv.w > xs[s]));
        c[s] = cc;
      }
      if (lane == 0) {
        ((int4*)cntBuf)[wv * 2 + 0] = make_int4(c[0], c[1], c[2], c[3]);
        ((int4*)cntBuf)[wv * 2 + 1] = make_int4(c[4], c[5], c[6], c[7]);
      }
    }

    // ---------- wave top-8 on exp values (monotone in x) ----------
    {
      float a0 = e.x, a1 = e.y, a2 = e.z, a3 = e.w;
      ceD(a0, a1); ceD(a2, a3); ceD(a0, a2); ceD(a1, a3); ceD(a1, a2); // sorted-4 desc
      float t[8];
      t[0] = a0; t[1] = a1; t[2] = a2; t[3] = a3;
      t[4] = __shfl_xor(a3, 1, 32); t[5] = __shfl_xor(a2, 1, 32);
      t[6] = __shfl_xor(a1, 1, 32); t[7] = __shfl_xor(a0, 1, 32);
      bitonic8(t);                                       // sorted-8 desc
      #pragma unroll
      for (int d = 2; d <= 16; d <<= 1) {                // top-8 of two sorted-8
        float q[8], u[8];
        #pragma unroll
        for (int i = 0; i < 8; ++i) q[i] = __shfl_xor(t[i], d, 32);
        #pragma unroll
        for (int i = 0; i < 8; ++i) u[i] = fmaxf(t[i], q[7 - i]); // bitonic top-8
        bitonic8(u);
        #pragma unroll
        for (int i = 0; i < 8; ++i) t[i] = u[i];
      }
      if (lane == 0) {
        ((float4*)topBuf)[wv * 2 + 0] = make_float4(t[0], t[1], t[2], t[3]);
        ((float4*)topBuf)[wv * 2 + 1] = make_float4(t[4], t[5], t[6], t[7]);
      }
    }
    __syncthreads();                                    // (C)

    float denom;
    {
      const float4 s0 = ((const float4*)sumBuf)[0];
      const float4 s1 = ((const float4*)sumBuf)[1];
      denom = ((s0.x + s0.y) + (s0.z + s0.w)) + ((s1.x + s1.y) + (s1.z + s1.w));
    }
    const float inv = 1.0f / denom;

    // ---------- wave 0: merge 64 candidates -> global top-8; m; ranks; scatter prep
    bool  doScat  = false;
    int   scatCol = 0;
    float scatVal = 0.0f;
    if (wv == 0) {
      float c0 = topBuf[lane], c1 = topBuf[32 + lane];
      ceD(c0, c1);                                      // sorted-2 desc
      float p0 = __shfl_xor(c0, 1, 32), p1 = __shfl_xor(c1, 1, 32);
      float s0c = c0, s1c = c1, s2c = p1, s3c = p0;     // bitonic-4
      ceD(s0c, s2c); ceD(s1c, s3c); ceD(s0c, s1c); ceD(s2c, s3c); // sorted-4 desc
      float g[8];
      g[0] = s0c; g[1] = s1c; g[2] = s2c; g[3] = s3c;
      g[4] = __shfl_xor(s3c, 2, 32); g[5] = __shfl_xor(s2c, 2, 32);
      g[6] = __shfl_xor(s1c, 2, 32); g[7] = __shfl_xor(s0c, 2, 32);
      bitonic8(g);                                      // sorted-8 desc
      #pragma unroll
      for (int d = 4; d <= 16; d <<= 1) {
        float q[8], u[8];
        #pragma unroll
        for (int i = 0; i < 8; ++i) q[i] = __shfl_xor(g[i], d, 32);
        #pragma unroll
        for (int i = 0; i < 8; ++i) u[i] = fmaxf(g[i], q[7 - i]);
        bitonic8(u);
        #pragma unroll
        for (int i = 0; i < 8; ++i) g[i] = u[i];
      }
      float tp[8];
      #pragma unroll
      for (int i = 0; i < 8; ++i) tp[i] = g[i] * inv;   // sorted probs
      float cum = 0.0f; int m = TOPK;
      #pragma unroll
      for (int i = 0; i < 8; ++i) { cum += tp[i]; if (m == TOPK && cum > TOPP_F) m = i + 1; }
      int r = 0;
      if (lane < 8) {                                   // rank of original column `lane`
        #pragma unroll
        for (int w2 = 0; w2 < NWAVES; ++w2) r += cntBuf[w2 * 8 + lane];
      }
      doScat  = (lane < 8) && (lane < m);
      scatCol = r;
      float sv = tp[0];
      #pragma unroll
      for (int i = 1; i < 8; ++i) sv = (lane == i) ? tp[i] : sv;
      scatVal = sv;
    }

    // ---------- output: -inf fill, then <=8 scatter stores ordered behind it ----
    float* orow = out + (size_t)row * N_COLS;
    ((float4*)orow)[tid] = make_float4(NEG_INF, NEG_INF, NEG_INF, NEG_INF);
    asm volatile("s_wait_storecnt 0x0" ::: "memory");   // fill has reached L2
    __syncthreads();                                    // (D)
    if (doScat) orow[scatCol] = scatVal;

    cur ^= 1;
  }
  #undef ISSUE_ASYNC
}

extern "C" void kernel_launch(void* const* d_in, const int* in_sizes, int n_in,
                              void* d_out, int out_size, void* d_ws, size_t ws_size,
                              hipStream_t stream) {
  (void)n_in; (void)d_ws; (void)ws_size; (void)out_size;
  const float* x = (const float*)d_in[0];
  float* out = (float*)d_out;
  const int rows = in_sizes[0] / N_COLS;               // 32*2048 = 65536
  if (rows <= 0) return;
  int grid = rows < 4096 ? rows : 4096;                 // ~16 rows/block: deep async pipeline
  topk_topp_kernel<<<grid, NBLK_THR, 0, stream>>>(x, out, rows);
}